// GraphMAE_58153857188005
// MI455X (gfx1250) — compile-verified
//
#include <hip/hip_runtime.h>

typedef __attribute__((ext_vector_type(2))) float v2f;
typedef __attribute__((ext_vector_type(8))) float v8f;
typedef __attribute__((ext_vector_type(4))) unsigned int u32x4;
typedef __attribute__((ext_vector_type(4))) int i32x4;
typedef __attribute__((ext_vector_type(8))) int i32x8;

// ---------------------------------------------------------------------------
// Elementwise helpers
// ---------------------------------------------------------------------------
__global__ void k_fill(float* __restrict__ p, float v, int n) {
    int i = blockIdx.x * blockDim.x + threadIdx.x;
    if (i < n) p[i] = v;
}

__global__ void k_deg_count(const long long* __restrict__ dst,
                            float* __restrict__ deg, int E) {
    int i = blockIdx.x * blockDim.x + threadIdx.x;
    if (i < E) atomicAdd(&deg[(int)dst[i]], 1.0f);
}

__global__ void k_rsqrt_inplace(float* __restrict__ p, int n) {
    int i = blockIdx.x * blockDim.x + threadIdx.x;
    if (i < n) {
        float d = p[i];
        p[i] = (d > 0.0f) ? rsqrtf(d) : 0.0f;
    }
}

template<int D>
__global__ void k_bias_init(float* __restrict__ out, const float* __restrict__ bias,
                            long long total) {
    long long i = (long long)blockIdx.x * blockDim.x + threadIdx.x;
    if (i < total) out[i] = bias[(int)(i & (D - 1))];
}

__global__ void k_relu(float* __restrict__ p, long long n) {
    long long i = (long long)blockIdx.x * blockDim.x + threadIdx.x;
    if (i < n) p[i] = fmaxf(p[i], 0.0f);
}

// ---------------------------------------------------------------------------
// FP32 WMMA GEMM:  C[M,NOUT] = act( (A .* mask?) @ W[K,NOUT] + bias? )
//
// Grid: (ceil(M/16/8), NOUT/16). Block = 256 threads = 8 waves.
// All 8 waves of a block share one 16-column stripe of W (blockIdx.y); the
// stripe (K x 16 f32) is DMA'd into LDS once by the Tensor Data Mover, then
// each wave computes one 16x16 tile with K/4 V_WMMA_F32_16X16X4_F32 ops
// (fully unrolled; compile-time strides -> 32-bit addressing).
// M must be a multiple of 16 (N = 50000 = 3125*16).
// ---------------------------------------------------------------------------
template<int K, int NOUT, bool HasMask, bool HasBias, bool DoRelu>
__global__ __launch_bounds__(256) void k_gemm_wmma(
    const float* __restrict__ A, const float* __restrict__ W,
    const float* __restrict__ bias, const float* __restrict__ mask,
    float* __restrict__ C, int M)
{
    __shared__ float Blds[K * 16];              // [K][16] row-major, 4-8 KB
    const int col0 = blockIdx.y << 4;

    // ---- TDM: DMA the K x 16 weight stripe into LDS (wave 0 issues) ----
    if (threadIdx.x < 32) {
        unsigned long long ga = (unsigned long long)(const void*)(W + col0);
        unsigned ldsOff = (unsigned)(unsigned long long)(const void*)&Blds[0];

        u32x4 g0;
        g0.x = 1u;                                            // count=1, no gather
        g0.y = ldsOff;                                        // lds_addr (bytes)
        g0.z = (unsigned)ga;                                  // global_addr[31:0]
        g0.w = ((unsigned)(ga >> 32) & 0x01FFFFFFu) | (2u << 30); // addr[56:32] | type=2

        i32x8 g1;
        g1[0] = (int)(2u << 16);                              // wg_mask=0, data_size=4B
        g1[1] = (int)(((unsigned)NOUT & 0xFFFFu) << 16);      // tensor_dim0[15:0]
        g1[2] = (int)(((unsigned)K & 0xFFFFu) << 16);         // dim0 hi=0 | tensor_dim1 lo=K
        g1[3] = (int)(16u << 16);                             // dim1 hi=0 | tile_dim0=16
        g1[4] = (int)((unsigned)K & 0xFFFFu);                 // tile_dim1=K | tile_dim2=0
        g1[5] = (int)(unsigned)NOUT;                          // tensor_dim0_stride lo32
        g1[6] = 0;                                            // stride hi | dim1_stride lo
        g1[7] = 0;

        i32x4 gz = {0, 0, 0, 0};
#if defined(__clang_major__) && (__clang_major__ >= 23)
        i32x8 gz8 = {0, 0, 0, 0, 0, 0, 0, 0};
        __builtin_amdgcn_tensor_load_to_lds(g0, g1, gz, gz, gz8, 0);
#else
        __builtin_amdgcn_tensor_load_to_lds(g0, g1, gz, gz, 0);
#endif
        __builtin_amdgcn_s_wait_tensorcnt(0);
    }
    __syncthreads();

    const int wave = threadIdx.x >> 5;
    const int lane = threadIdx.x & 31;
    const int tm = blockIdx.x * (blockDim.x >> 5) + wave;
    const int tilesM = (M + 15) >> 4;
    if (tm >= tilesM) return;                  // whole-wave exit after barrier

    const int half = lane >> 4;                // 0 -> k pair {0,1}, 1 -> {2,3}
    const int l16  = lane & 15;
    const int row0 = tm << 4;
    const int arow = row0 + l16;

    float mscale = 1.0f;
    if (HasMask) mscale = mask[arow];

    const v2f* __restrict__ Arow = (const v2f*)(A + (size_t)arow * K);

    v8f acc = {};
#pragma unroll
    for (int k0 = 0; k0 < K; k0 += 4) {
        v2f a = Arow[(k0 >> 1) + half];        // A[arow][k0+2h .. k0+2h+1]
        if (HasMask) { a.x *= mscale; a.y *= mscale; }
        const int ka = k0 + (half << 1);
        v2f b;
        b.x = Blds[ka * 16 + l16];             // W[ka  ][col0+l16]
        b.y = Blds[ka * 16 + 16 + l16];        // W[ka+1][col0+l16]
        // (neg_a, A, neg_b, B, c_mod, C, reuse_a, reuse_b)
        acc = __builtin_amdgcn_wmma_f32_16x16x4_f32(
            false, a, false, b, (short)0, acc, false, false);
    }

    const int gcol = col0 + l16;
    float bb = 0.0f;
    if (HasBias) bb = bias[gcol];
#pragma unroll
    for (int i = 0; i < 8; ++i) {
        const int orow = row0 + i + (half << 3);
        float v = acc[i] + bb;
        if (DoRelu) v = fmaxf(v, 0.0f);
        C[(size_t)orow * NOUT + gcol] = v;
    }
}

// ---------------------------------------------------------------------------
// Edge scatter: out[dst] += H[src] * (dinv[src]*dinv[dst])
// One wave per edge; ids >= E are self-loops. Lane-vectorized D/32 floats.
// ---------------------------------------------------------------------------
template<int D>
__global__ void k_scatter(const float* __restrict__ H,
                          const long long* __restrict__ src,
                          const long long* __restrict__ dst,
                          const float* __restrict__ dinv,
                          float* __restrict__ out,
                          int E, int Nn) {
    const int wave = threadIdx.x >> 5;
    const int lane = threadIdx.x & 31;
    const long long e = (long long)blockIdx.x * (blockDim.x >> 5) + wave;
    if (e >= (long long)E + Nn) return;

    int s, d;
    if (e < E) { s = (int)src[e]; d = (int)dst[e]; }
    else       { s = d = (int)(e - E); }

    const float nrm = dinv[s] * dinv[d];
    constexpr int V = D / 32;                  // 4 (D=128) or 2 (D=64)
    const float* __restrict__ hp = H + (size_t)s * D + lane * V;
    float* __restrict__ op = out + (size_t)d * D + lane * V;

    float v[V];
#pragma unroll
    for (int i = 0; i < V; ++i) v[i] = hp[i];  // merges into b128/b64 load
#pragma unroll
    for (int i = 0; i < V; ++i) atomicAdd(&op[i], v[i] * nrm);
}

// ---------------------------------------------------------------------------
extern "C" void kernel_launch(void* const* d_in, const int* in_sizes, int n_in,
                              void* d_out, int out_size, void* d_ws, size_t ws_size,
                              hipStream_t stream) {
    const float*      x     = (const float*)d_in[0];
    const long long*  ei    = (const long long*)d_in[1];  // [2, E] int64
    const float*      maskv = (const float*)d_in[2];      // [N, 1]
    const float*      W_e1  = (const float*)d_in[3];
    const float*      b_e1  = (const float*)d_in[4];
    const float*      W_e2  = (const float*)d_in[5];
    const float*      b_e2  = (const float*)d_in[6];
    const float*      W_d1  = (const float*)d_in[7];
    const float*      b_d1  = (const float*)d_in[8];
    const float*      W_d2  = (const float*)d_in[9];
    const float*      b_d2  = (const float*)d_in[10];
    (void)n_in; (void)out_size; (void)ws_size;

    constexpr int IN_DIM = 128, HID = 128, LAT = 64;
    const int Nn = in_sizes[2];          // mask_vector is [N,1]
    const int E  = in_sizes[1] / 2;
    const long long* src = ei;
    const long long* dst = ei + E;

    // workspace layout
    auto align256 = [](size_t v) { return (v + 255) & ~(size_t)255; };
    char* ws = (char*)d_ws;
    size_t off = 0;
    float* dinv = (float*)(ws + off); off += align256((size_t)Nn * 4);
    float* bufA = (float*)(ws + off); off += align256((size_t)Nn * HID * 4); // H0 -> T2
    float* bufB = (float*)(ws + off); off += align256((size_t)Nn * HID * 4); // h  -> x_hat

    float* xrecon = (float*)d_out;                       // [N,128]
    float* z      = (float*)d_out + (size_t)Nn * IN_DIM; // [N,64]

    const int TPB = 256;
    const int WPB = TPB / 32;
    const int tilesM = (Nn + 15) / 16;
    const int gx = (tilesM + WPB - 1) / WPB;
    const long long edges = (long long)E + Nn;
    const int egrid = (int)((edges + WPB - 1) / WPB);

    // ---- GCN normalization: deg (with self loops) -> rsqrt ----
    k_fill<<<(Nn + TPB - 1) / TPB, TPB, 0, stream>>>(dinv, 1.0f, Nn);
    k_deg_count<<<(E + TPB - 1) / TPB, TPB, 0, stream>>>(dst, dinv, E);
    k_rsqrt_inplace<<<(Nn + TPB - 1) / TPB, TPB, 0, stream>>>(dinv, Nn);

    // ---- Encoder layer 1: H0 = (x .* mask) @ W_e1 ----
    k_gemm_wmma<IN_DIM, HID, true, false, false>
        <<<dim3(gx, HID / 16), TPB, 0, stream>>>(x, W_e1, nullptr, maskv, bufA, Nn);
    // h = relu(scatter(H0) + b_e1)
    {
        long long tot = (long long)Nn * HID;
        k_bias_init<HID><<<(int)((tot + TPB - 1) / TPB), TPB, 0, stream>>>(bufB, b_e1, tot);
        k_scatter<HID><<<egrid, TPB, 0, stream>>>(bufA, src, dst, dinv, bufB, E, Nn);
        k_relu<<<(int)((tot + TPB - 1) / TPB), TPB, 0, stream>>>(bufB, tot);
    }

    // ---- Encoder layer 2: T2 = h @ W_e2 ----
    k_gemm_wmma<HID, LAT, false, false, false>
        <<<dim3(gx, LAT / 16), TPB, 0, stream>>>(bufB, W_e2, nullptr, nullptr, bufA, Nn);
    // z = scatter(T2) + b_e2   (written straight into d_out tail)
    {
        long long tot = (long long)Nn * LAT;
        k_bias_init<LAT><<<(int)((tot + TPB - 1) / TPB), TPB, 0, stream>>>(z, b_e2, tot);
        k_scatter<LAT><<<egrid, TPB, 0, stream>>>(bufA, src, dst, dinv, z, E, Nn);
    }

    // ---- Decoder: x_hat = relu(z @ W_d1 + b_d1) ----
    k_gemm_wmma<LAT, HID, false, true, true>
        <<<dim3(gx, HID / 16), TPB, 0, stream>>>(z, W_d1, b_d1, nullptr, bufB, Nn);
    // ---- x_recon = x_hat @ W_d2 + b_d2 ----
    k_gemm_wmma<HID, IN_DIM, false, true, false>
        <<<dim3(gx, IN_DIM / 16), TPB, 0, stream>>>(bufB, W_d2, b_d2, nullptr, xrecon, Nn);
}